// ConvexComboReward_4595615007164
// MI455X (gfx1250) — compile-verified
//
#include <hip/hip_runtime.h>
#include <math.h>

typedef __attribute__((ext_vector_type(2))) float v2f;
typedef __attribute__((ext_vector_type(8))) float v8f;
typedef int v4i __attribute__((vector_size(4 * sizeof(int))));  // matches builtin's param type

#define F     128
#define NROWS 2048
#define MROWS 2048
#define EPSW  1e-4f

#define NB    8     // n-rows per workgroup in main kernel
#define TM    128   // m-tile rows staged per LDS buffer
#define TPB   128   // threads per block (4 waves)
#define SBST  132   // padded LDS row stride (floats): 16B-aligned, kills bank conflicts
#define NT    (MROWS / TM)          // 16 tiles
#define ISSUES ((TM * F * 4) / (TPB * 16))  // 32 async b128 issues per thread per tile

#define AS1 __attribute__((address_space(1)))
#define AS3 __attribute__((address_space(3)))

#if __has_builtin(__builtin_amdgcn_global_load_async_to_lds_b128)
#define HAVE_ASYNC_LDS 1
#if __has_builtin(__builtin_amdgcn_s_wait_asynccnt)
#define WAIT_ASYNC(n) __builtin_amdgcn_s_wait_asynccnt(n)
#else
#define WAIT_ASYNC(n) asm volatile("s_wait_asynccnt %0" :: "i"(n) : "memory")
#endif
#endif

// ---------------------------------------------------------------------------
// Kernel 1: Y = X @ W1, X = [embeds ; base_embeds] (4096 x 128), exact fp32
// via V_WMMA_F32_16X16X4_F32, one wave per 16x16 output tile. (Unchanged:
// round-1 codegen confirmed v_wmma pipelined against partial loadcnt waits.)
// ---------------------------------------------------------------------------
__global__ __launch_bounds__(32)
void mlp_layer1_wmma(const float* __restrict__ embeds,
                     const float* __restrict__ base_embeds,
                     const float* __restrict__ W1,   // [F][F] row-major (in,out)
                     float* __restrict__ Y) {        // [4096][F]
  const int lane = threadIdx.x;
  const int half = lane >> 4;
  const int l16  = lane & 15;
  const int colTile = blockIdx.x;   // 0..7
  const int rowTile = blockIdx.y;   // 0..255

  const int row = rowTile * 16 + l16;
  const float* xrow = (row < NROWS) ? (embeds + (size_t)row * F)
                                    : (base_embeds + (size_t)(row - NROWS) * F);
  const int col = colTile * 16 + l16;

  v8f acc = {};
  #pragma unroll 4
  for (int k0 = 0; k0 < F; k0 += 4) {
    const int ka = k0 + half * 2;
    v2f a, b;
    a.x = xrow[ka];
    a.y = xrow[ka + 1];
    b.x = W1[(size_t)ka * F + col];
    b.y = W1[(size_t)(ka + 1) * F + col];
    acc = __builtin_amdgcn_wmma_f32_16x16x4_f32(false, a, false, b,
                                                (short)0, acc, false, false);
  }

  const int orow0 = rowTile * 16 + half * 8;
  #pragma unroll
  for (int r = 0; r < 8; ++r)
    Y[(size_t)(orow0 + r) * F + col] = acc[r];
}

// ---------------------------------------------------------------------------
// Kernel 2: rewards[n] = sum_m w(n,m)*r[m] / sum_m w(n,m),
//           w(n,m) = 1/(|sum_g ELU(A[n][g]-B[m][g])*w2[g]| + eps)
// NB=8 A-rows pinned in LDS; B streamed in double-buffered 64KB tiles via
// GLOBAL_LOAD_ASYNC_TO_LDS_B128 (ASYNCcnt), overlapping copy with compute.
// Thread t owns m = m0 + t for all 8 n.
// ---------------------------------------------------------------------------
__global__ __launch_bounds__(TPB)
void reward_kernel(const float* __restrict__ Aemb,         // [NROWS][F]
                   const float* __restrict__ Bemb,         // [MROWS][F]
                   const float* __restrict__ w2,           // [F]
                   const float* __restrict__ base_rewards, // [MROWS]
                   float* __restrict__ out) {              // [NROWS]
  extern __shared__ float smem[];
  float* sA  = smem;                  // NB*F    = 1024 floats
  float* sw  = sA + NB * F;           // F       = 128
  float* sRn = sw + F;                // TPB     = 128
  float* sRd = sRn + TPB;             // TPB     = 128
  float* sB0 = sRd + TPB;             // TM*SBST = 16896
  float* sB1 = sB0 + TM * SBST;       // TM*SBST = 16896   (~138 KB total)

  const int t  = threadIdx.x;
  const int n0 = blockIdx.x * NB;

  for (int i = t; i < NB * F; i += TPB) sA[i] = Aemb[(size_t)n0 * F + i];
  for (int i = t; i < F; i += TPB)      sw[i] = w2[i];

  float num[NB], den[NB];
  #pragma unroll
  for (int n = 0; n < NB; ++n) { num[n] = 0.f; den[n] = 0.f; }

#if defined(HAVE_ASYNC_LDS)
  // ---- async double-buffered pipeline ------------------------------------
  // issue one tile: TM x F floats as per-lane 16B async copies, coalesced in
  // global, padded rows (SBST) in LDS. 16B LDS alignment: (row*528+col4*16).
  #define ISSUE_TILE(m0_, dst_)                                              \
    do {                                                                     \
      _Pragma("unroll")                                                      \
      for (int i = 0; i < ISSUES; ++i) {                                     \
        const int e4   = t + TPB * i;   /* float4 index in tile */           \
        const int rowi = e4 >> 5;       /* / (F/4) */                        \
        const int col4 = e4 & 31;                                            \
        __builtin_amdgcn_global_load_async_to_lds_b128(                      \
            (AS1 v4i*)(Bemb + (size_t)(m0_)*F + (size_t)e4 * 4),             \
            (AS3 v4i*)((dst_) + rowi * SBST + col4 * 4), 0, 0);              \
      }                                                                      \
    } while (0)

  ISSUE_TILE(0, sB0);
  for (int it = 0; it < NT; ++it) {
    float* cur = (it & 1) ? sB1 : sB0;
    float* nxt = (it & 1) ? sB0 : sB1;
    if (it + 1 < NT) {
      ISSUE_TILE((it + 1) * TM, nxt);
      WAIT_ASYNC(ISSUES);   // oldest ISSUES (current tile) complete, in order
    } else {
      WAIT_ASYNC(0);
    }
    __syncthreads();        // all waves' copies for 'cur' visible
#else
  // ---- fallback: synchronous staging -------------------------------------
  for (int it = 0; it < NT; ++it) {
    float* cur = sB0;
    __syncthreads();
    const float2* gsrc = (const float2*)(Bemb + (size_t)it * TM * F);
    #pragma unroll 4
    for (int i = 0; i < (TM * F) / (2 * TPB); ++i) {
      const int e2   = t + TPB * i;
      const int rowi = e2 >> 6;
      const int col2 = e2 & 63;
      const float2 v = gsrc[e2];
      cur[rowi * SBST + col2 * 2]     = v.x;
      cur[rowi * SBST + col2 * 2 + 1] = v.y;
    }
    __syncthreads();
#endif
    // ---- compute on tile 'cur' -------------------------------------------
    const int   m  = it * TM + t;
    const float rm = base_rewards[m];
    const float* bRow = cur + t * SBST;

    float dot[NB];
    #pragma unroll
    for (int n = 0; n < NB; ++n) dot[n] = 0.f;

    for (int g = 0; g < F; g += 4) {
      const float4 b4 = *(const float4*)(bRow + g);
      const float4 w4 = *(const float4*)(sw + g);
      #pragma unroll
      for (int n = 0; n < NB; ++n) {
        const float4 a4 = *(const float4*)(sA + n * F + g);
        float x, h;
        x = a4.x - b4.x; h = (x > 0.f) ? x : (__expf(x) - 1.f); dot[n] = fmaf(h, w4.x, dot[n]);
        x = a4.y - b4.y; h = (x > 0.f) ? x : (__expf(x) - 1.f); dot[n] = fmaf(h, w4.y, dot[n]);
        x = a4.z - b4.z; h = (x > 0.f) ? x : (__expf(x) - 1.f); dot[n] = fmaf(h, w4.z, dot[n]);
        x = a4.w - b4.w; h = (x > 0.f) ? x : (__expf(x) - 1.f); dot[n] = fmaf(h, w4.w, dot[n]);
      }
    }

    #pragma unroll
    for (int n = 0; n < NB; ++n) {
      const float d = fabsf(dot[n]);
      const float w = 1.f / (d + EPSW);
      den[n] += w;
      num[n] = fmaf(w, rm, num[n]);
    }

#if defined(HAVE_ASYNC_LDS)
    __syncthreads();   // 'cur' fully consumed before it is refilled at it+2
  }
#else
  }
#endif

  // per-n reduction over the TPB threads (each owned a disjoint set of m)
  #pragma unroll 1
  for (int n = 0; n < NB; ++n) {
    __syncthreads();
    sRn[t] = num[n];
    sRd[t] = den[n];
    __syncthreads();
    for (int s = TPB / 2; s > 0; s >>= 1) {
      if (t < s) { sRn[t] += sRn[t + s]; sRd[t] += sRd[t + s]; }
      __syncthreads();
    }
    if (t == 0) out[n0 + n] = sRn[0] / sRd[0];
  }
}

// ---------------------------------------------------------------------------
extern "C" void kernel_launch(void* const* d_in, const int* in_sizes, int n_in,
                              void* d_out, int out_size, void* d_ws, size_t ws_size,
                              hipStream_t stream) {
  const float* embeds       = (const float*)d_in[0]; // [2048][128]
  const float* base_embeds  = (const float*)d_in[1]; // [2048][128]
  const float* base_rewards = (const float*)d_in[2]; // [2048]
  const float* W1           = (const float*)d_in[3]; // [128][128]
  const float* w2           = (const float*)d_in[4]; // [128]
  float* out = (float*)d_out;                        // [2048]

  float* Y    = (float*)d_ws;              // [4096][128] fp32 = 2 MB scratch
  float* Aemb = Y;                         // embeds @ W1
  float* Bemb = Y + (size_t)NROWS * F;     // base_embeds @ W1

  dim3 g1(F / 16, (NROWS + MROWS) / 16);   // (8, 256) waves of 32
  mlp_layer1_wmma<<<g1, 32, 0, stream>>>(embeds, base_embeds, W1, Y);

  const size_t smem_bytes =
      (size_t)(NB * F + F + 2 * TPB + 2 * TM * SBST) * sizeof(float); // ~138 KB
  (void)hipFuncSetAttribute((const void*)reward_kernel,
                            hipFuncAttributeMaxDynamicSharedMemorySize,
                            (int)smem_bytes);
  reward_kernel<<<NROWS / NB, TPB, smem_bytes, stream>>>(
      Aemb, Bemb, w2, base_rewards, out);
}